// RBFInterpolator_34900904247904
// MI455X (gfx1250) — compile-verified
//
#include <hip/hip_runtime.h>
#include <hip/hip_bf16.h>

typedef __attribute__((ext_vector_type(2))) float v2f;
typedef __attribute__((ext_vector_type(8))) float v8f;

#define Q_TOT   32768
#define N_PTS   1024
#define C_CH    32
#define TILE_Q  16
#define WAVES_PER_BLOCK 4
#define THREADS (WAVES_PER_BLOCK * 32)

// One wave computes one 16-query tile:
//   - pass A: row sums of exp(-r^2) over all N (lane -> q = lane%16, n-parity = lane/16)
//   - pass B: rebuild normalized K in WMMA A-fragment layout, store it to
//     q_kernel_dist, and accumulate K @ w with V_WMMA_F32_16X16X4_F32.
__global__ __launch_bounds__(THREADS) void rbf_interp_kernel(
    const float* __restrict__ q_pos,    // [Q,3]
    const float* __restrict__ data_pos, // [N,3]
    const float* __restrict__ w,        // [N,C]
    const float* __restrict__ cmat,     // [4,C]
    float* __restrict__ out_wval,       // [Q,C]
    float* __restrict__ out_kdist,      // [Q,N]
    float* __restrict__ out_poly)       // [Q,4]
{
    __shared__ float lds_dp[N_PTS * 3];                    // 12 KB, shared by block
    __shared__ float lds_q[WAVES_PER_BLOCK][TILE_Q][3];    // per-wave query tile

    const int tid  = threadIdx.x;
    const int wave = tid >> 5;
    const int lane = tid & 31;
    const int l16  = lane & 15;
    const int hi   = lane >> 4;   // 0: lanes 0-15, 1: lanes 16-31

    // Stage data_pos into LDS cooperatively.
    for (int i = tid; i < N_PTS * 3; i += THREADS) lds_dp[i] = data_pos[i];

    const int qbase = (blockIdx.x * WAVES_PER_BLOCK + wave) * TILE_Q;
    const int q     = qbase + l16;              // this lane's query row for K
    const float qx = q_pos[q * 3 + 0];
    const float qy = q_pos[q * 3 + 1];
    const float qz = q_pos[q * 3 + 2];
    if (hi == 0) {
        lds_q[wave][l16][0] = qx;
        lds_q[wave][l16][1] = qy;
        lds_q[wave][l16][2] = qz;
    }
    __syncthreads();

    // ---- Pass A: row sums (lane covers n = 2*i + hi) ----
    float psum = 0.0f;
    for (int i = 0; i < N_PTS / 2; ++i) {
        const int n = 2 * i + hi;
        const float dx = qx - lds_dp[n * 3 + 0];
        const float dy = qy - lds_dp[n * 3 + 1];
        const float dz = qz - lds_dp[n * 3 + 2];
        psum += __expf(-(dx * dx + dy * dy + dz * dz));
    }
    // lanes l and l+16 hold the two halves of row q = l%16
    const float rsum = psum + __shfl_xor(psum, 16, 32);
    const float inv  = 1.0f / rsum;

    // ---- Pass B: WMMA accumulation over N in K=4 chunks ----
    // A fragment layout (fp32 16x4): VGPR0 = K{0 | 2}, VGPR1 = K{1 | 3}
    // across lane halves; this lane supplies K = 2*hi and 2*hi+1 for row q.
    v8f acc0 = {0.f, 0.f, 0.f, 0.f, 0.f, 0.f, 0.f, 0.f};
    v8f acc1 = {0.f, 0.f, 0.f, 0.f, 0.f, 0.f, 0.f, 0.f};
    for (int n0 = 0; n0 < N_PTS; n0 += 4) {
        const int na = n0 + 2 * hi;

        float dx = qx - lds_dp[na * 3 + 0];
        float dy = qy - lds_dp[na * 3 + 1];
        float dz = qz - lds_dp[na * 3 + 2];
        const float k0 = __expf(-(dx * dx + dy * dy + dz * dz)) * inv;

        dx = qx - lds_dp[(na + 1) * 3 + 0];
        dy = qy - lds_dp[(na + 1) * 3 + 1];
        dz = qz - lds_dp[(na + 1) * 3 + 2];
        const float k1 = __expf(-(dx * dx + dy * dy + dz * dz)) * inv;

        // Normalized kernel row -> global output (8B contiguous per lane).
        v2f kk; kk.x = k0; kk.y = k1;
        *(v2f*)(out_kdist + (size_t)q * N_PTS + na) = kk;

        v2f a; a.x = k0; a.y = k1;

        // B fragment (fp32 4x16), mirrored layout: VGPR0 = row K{0|2}, VGPR1 = row K{1|3}.
        v2f b0, b1;
        b0.x = w[(size_t)na * C_CH + l16];
        b0.y = w[(size_t)(na + 1) * C_CH + l16];
        b1.x = w[(size_t)na * C_CH + 16 + l16];
        b1.y = w[(size_t)(na + 1) * C_CH + 16 + l16];

        acc0 = __builtin_amdgcn_wmma_f32_16x16x4_f32(false, a, false, b0,
                                                     (short)0, acc0, false, false);
        acc1 = __builtin_amdgcn_wmma_f32_16x16x4_f32(false, a, false, b1,
                                                     (short)0, acc1, false, false);
    }

    // ---- Poly basis output: [1, x, y, z] per query (lanes 0-15) ----
    if (hi == 0) {
        float4 pv; pv.x = 1.0f; pv.y = qx; pv.z = qy; pv.w = qz;
        *(float4*)(out_poly + (size_t)q * 4) = pv;
    }

    // ---- Epilogue: add c_val and store w_val ----
    // C/D layout (16x16 f32): VGPR j, lanes 0-15 -> row j; lanes 16-31 -> row j+8.
    float c0[4], c1[4];
#pragma unroll
    for (int p = 0; p < 4; ++p) {
        c0[p] = cmat[p * C_CH + l16];
        c1[p] = cmat[p * C_CH + 16 + l16];
    }
#pragma unroll
    for (int j = 0; j < 8; ++j) {
        const int row = j + 8 * hi;   // row within tile
        const float rx = lds_q[wave][row][0];
        const float ry = lds_q[wave][row][1];
        const float rz = lds_q[wave][row][2];
        const float cv0 = c0[0] + rx * c0[1] + ry * c0[2] + rz * c0[3];
        const float cv1 = c1[0] + rx * c1[1] + ry * c1[2] + rz * c1[3];
        const int gq = qbase + row;
        out_wval[(size_t)gq * C_CH + l16]      = acc0[j] + cv0;
        out_wval[(size_t)gq * C_CH + 16 + l16] = acc1[j] + cv1;
    }
}

extern "C" void kernel_launch(void* const* d_in, const int* in_sizes, int n_in,
                              void* d_out, int out_size, void* d_ws, size_t ws_size,
                              hipStream_t stream) {
    (void)in_sizes; (void)n_in; (void)d_ws; (void)ws_size; (void)out_size;

    const float* q_pos    = (const float*)d_in[0];
    const float* data_pos = (const float*)d_in[1];
    const float* w        = (const float*)d_in[2];
    const float* cmat     = (const float*)d_in[3];
    // d_in[4] = poly_power (int32 [4,3]) — fixed degree<=1 basis, folded into the kernel.

    float* out = (float*)d_out;
    float* out_wval  = out;                                        // [Q,C]
    float* out_kdist = out + (size_t)Q_TOT * C_CH;                 // [Q,N]
    float* out_poly  = out + (size_t)Q_TOT * C_CH
                           + (size_t)Q_TOT * N_PTS;                // [Q,4]

    const int tiles  = Q_TOT / TILE_Q;            // 2048
    const int blocks = tiles / WAVES_PER_BLOCK;   // 512
    rbf_interp_kernel<<<blocks, THREADS, 0, stream>>>(
        q_pos, data_pos, w, cmat, out_wval, out_kdist, out_poly);
}